// SimpleHGNLayer_85031762526242
// MI455X (gfx1250) — compile-verified
//
#include <hip/hip_runtime.h>
#include <hip/hip_bf16.h>

typedef __attribute__((ext_vector_type(2))) float v2f;
typedef __attribute__((ext_vector_type(8))) float v8f;

#define N_NODES 50000
#define N_EDGES 1600000
#define IN_HIDDEN 256
#define OUT_COLS 128      // NHEAD * OUT_HIDDEN
#define NHEAD 4
#define OUT_HIDDEN 32
#define EDGE_HIDDEN 64
#define N_ETYPES 8
#define NEG_SLOPE 0.2f

// ---------- helpers: order-preserving float<->uint encoding for atomic max ----------
__device__ __forceinline__ unsigned enc_f32(float f) {
    unsigned u = __float_as_uint(f);
    return (u & 0x80000000u) ? ~u : (u | 0x80000000u);
}
__device__ __forceinline__ float dec_f32(unsigned u) {
    unsigned b = (u & 0x80000000u) ? (u & 0x7FFFFFFFu) : ~u;
    return __uint_as_float(b);
}

// ---------- kernel 0: init global-max cell to enc(-inf) ----------
__global__ void init_gmax_kernel(unsigned* gmax) {
    *gmax = enc_f32(-INFINITY);
}

// ---------- kernel 1: h = x @ W  (f32 WMMA), with NaN->0 on store ----------
// One wave per 16-row strip; each wave computes all 8 column tiles (16 cols each),
// reusing the A fragment across tiles. K loop: 256 in steps of 4.
__global__ void __launch_bounds__(256)
gemm_h_kernel(const float* __restrict__ x, const float* __restrict__ W,
              float* __restrict__ h) {
    const int lane  = threadIdx.x & 31;
    const int wave  = threadIdx.x >> 5;
    const int strip = blockIdx.x * (blockDim.x >> 5) + wave;   // 16-row strip
    if (strip * 16 >= N_NODES) return;   // wave-uniform: EXEC stays all-ones for WMMA
    const int row0  = strip * 16;

    const int mrow  = lane & 15;        // M index (A) / N index (B)
    const int khalf = (lane >> 4) << 1; // 0 for lanes 0-15, 2 for lanes 16-31

    v8f acc[8] = {};

    const float* xrow = x + (size_t)(row0 + mrow) * IN_HIDDEN + khalf;
    for (int k = 0; k < IN_HIDDEN; k += 4) {
        // A fragment: lane holds x[row0+mrow, k+khalf] and x[row0+mrow, k+khalf+1]
        v2f a = *(const v2f*)(xrow + k);
#pragma unroll
        for (int t = 0; t < 8; ++t) {
            // B fragment: lane holds W[k+khalf, col] and W[k+khalf+1, col]
            const float* wp = W + (size_t)(k + khalf) * OUT_COLS + t * 16 + mrow;
            v2f b;
            b.x = wp[0];
            b.y = wp[OUT_COLS];
            acc[t] = __builtin_amdgcn_wmma_f32_16x16x4_f32(
                false, a, false, b, (short)0, acc[t], false, false);
        }
    }

    // store D: vgpr i, lane L -> (M = i + 8*(L>=16), N = L&15)
    const int mbase = (lane >> 4) << 3;
#pragma unroll
    for (int t = 0; t < 8; ++t) {
#pragma unroll
        for (int i = 0; i < 8; ++i) {
            float v = acc[t][i];
            v = (v == v) ? v : 0.0f;   // jnp.where(isnan(h), 0, h)
            h[(size_t)(row0 + mbase + i) * OUT_COLS + t * 16 + mrow] = v;
        }
    }
}

// ---------- kernel 2: he[t,head] = sum_d a_e[head,d] * (edge_emb @ W_e)[t, head*64+d] ----------
__global__ void etype_logits_kernel(const float* __restrict__ edge_emb,
                                    const float* __restrict__ W_e,
                                    const float* __restrict__ a_e,
                                    float* __restrict__ he) {
    const int tid = threadIdx.x;          // 32 threads: (t, head)
    if (tid >= N_ETYPES * NHEAD) return;
    const int t = tid >> 2, head = tid & 3;
    float s = 0.0f;
    for (int d = 0; d < EDGE_HIDDEN; ++d) {
        float ev = 0.0f;
        for (int k = 0; k < EDGE_HIDDEN; ++k)
            ev = fmaf(edge_emb[t * EDGE_HIDDEN + k],
                      W_e[k * (EDGE_HIDDEN * NHEAD) + head * EDGE_HIDDEN + d], ev);
        s = fmaf(a_e[head * EDGE_HIDDEN + d], ev, s);
    }
    he[tid] = s;
}

// ---------- kernel 3: hl/hr per (node, head) ----------
__global__ void __launch_bounds__(256)
node_logits_kernel(const float* __restrict__ h,
                   const float* __restrict__ a_l, const float* __restrict__ a_r,
                   float* __restrict__ hl, float* __restrict__ hr) {
    const int idx = blockIdx.x * blockDim.x + threadIdx.x;
    if (idx >= N_NODES * NHEAD) return;
    const int node = idx >> 2, head = idx & 3;
    const float* hp = h + (size_t)node * OUT_COLS + head * OUT_HIDDEN;
    const float* al = a_l + head * OUT_HIDDEN;
    const float* ar = a_r + head * OUT_HIDDEN;
    float sl = 0.0f, sr = 0.0f;
#pragma unroll
    for (int d = 0; d < OUT_HIDDEN; ++d) {
        float v = hp[d];
        sl = fmaf(al[d], v, sl);
        sr = fmaf(ar[d], v, sr);
    }
    hl[idx] = sl;
    hr[idx] = sr;
}

// ---------- kernel 4: per-edge scores + global max ----------
__global__ void __launch_bounds__(256)
scores_kernel(const int* __restrict__ edge,
              const float* __restrict__ hl, const float* __restrict__ hr,
              const float* __restrict__ he,
              float* __restrict__ scores, unsigned* __restrict__ gmax) {
    const int e = blockIdx.x * blockDim.x + threadIdx.x;
    float lmax = -INFINITY;
    if (e < N_EDGES) {
        const int src = edge[e];
        const int trg = edge[N_EDGES + e];
        const int et  = edge[2 * N_EDGES + e];
#pragma unroll
        for (int hh = 0; hh < NHEAD; ++hh) {
            float s = hl[src * NHEAD + hh] + hr[trg * NHEAD + hh] + he[et * NHEAD + hh];
            s = (s > 0.0f) ? s : NEG_SLOPE * s;   // leaky_relu
            scores[(size_t)e * NHEAD + hh] = s;
            lmax = fmaxf(lmax, s);
        }
    }
    __shared__ float red[256];
    red[threadIdx.x] = lmax;
    __syncthreads();
    for (int off = 128; off > 0; off >>= 1) {
        if (threadIdx.x < off)
            red[threadIdx.x] = fmaxf(red[threadIdx.x], red[threadIdx.x + off]);
        __syncthreads();
    }
    if (threadIdx.x == 0) atomicMax(gmax, enc_f32(red[0]));
}

// ---------- kernel 5: exp(scores - max) + segment-sum denominator ----------
__global__ void __launch_bounds__(256)
expsum_kernel(const int* __restrict__ edge, float* __restrict__ scores,
              const unsigned* __restrict__ gmax, float* __restrict__ denom) {
    const int e = blockIdx.x * blockDim.x + threadIdx.x;
    if (e >= N_EDGES) return;
    const float smax = dec_f32(*gmax);
    const int trg = edge[N_EDGES + e];
#pragma unroll
    for (int hh = 0; hh < NHEAD; ++hh) {
        float es = __expf(scores[(size_t)e * NHEAD + hh] - smax);
        scores[(size_t)e * NHEAD + hh] = es;   // reuse buffer for exp_scores
        atomicAdd(&denom[trg * NHEAD + hh], es);
    }
}

// ---------- kernel 6: attn + weighted scatter-add aggregation ----------
// One wave (32 lanes) per edge; lane l handles column l of each head's 32-wide slice
// => fully coalesced 128B gathers from h[src] and 128B atomic adds into out[trg].
__global__ void __launch_bounds__(256)
aggregate_kernel(const int* __restrict__ edge, const float* __restrict__ h,
                 const float* __restrict__ es, const float* __restrict__ denom,
                 float* __restrict__ out, float* __restrict__ attn_out) {
    const int gid  = blockIdx.x * blockDim.x + threadIdx.x;
    const int e    = gid >> 5;
    const int lane = gid & 31;
    if (e >= N_EDGES) return;
    const int src = edge[e];
    const int trg = edge[N_EDGES + e];
    float a[NHEAD];
#pragma unroll
    for (int k = 0; k < NHEAD; ++k)
        a[k] = es[(size_t)e * NHEAD + k] / (denom[trg * NHEAD + k] + 1e-16f);
    if (lane < NHEAD) attn_out[(size_t)e * NHEAD + lane] = a[lane];
#pragma unroll
    for (int k = 0; k < NHEAD; ++k) {
        float m = h[(size_t)src * OUT_COLS + k * OUT_HIDDEN + lane] * a[k];
        atomicAdd(&out[(size_t)trg * OUT_COLS + k * OUT_HIDDEN + lane], m);
    }
}

extern "C" void kernel_launch(void* const* d_in, const int* in_sizes, int n_in,
                              void* d_out, int out_size, void* d_ws, size_t ws_size,
                              hipStream_t stream) {
    const int*   edge     = (const int*)  d_in[0];   // [3, E]
    const float* x        = (const float*)d_in[1];   // [N, 256]
    const float* edge_emb = (const float*)d_in[2];   // [8, 64]
    const float* W        = (const float*)d_in[3];   // [256, 128]
    const float* W_e      = (const float*)d_in[4];   // [64, 256]
    const float* a_l      = (const float*)d_in[5];   // [1, 4, 32]
    const float* a_r      = (const float*)d_in[6];   // [1, 4, 32]
    const float* a_e      = (const float*)d_in[7];   // [1, 4, 64]

    float* out      = (float*)d_out;                       // [N, 4, 32]
    float* attn_out = out + (size_t)N_NODES * OUT_COLS;    // [E, 4, 1]

    // workspace layout
    char* ws = (char*)d_ws;
    float* h      = (float*)ws;  ws += (size_t)N_NODES * OUT_COLS * sizeof(float); // 25.6 MB
    float* scores = (float*)ws;  ws += (size_t)N_EDGES * NHEAD   * sizeof(float);  // 25.6 MB
    float* hl     = (float*)ws;  ws += (size_t)N_NODES * NHEAD   * sizeof(float);
    float* hr     = (float*)ws;  ws += (size_t)N_NODES * NHEAD   * sizeof(float);
    float* denom  = (float*)ws;  ws += (size_t)N_NODES * NHEAD   * sizeof(float);
    float* he     = (float*)ws;  ws += (size_t)N_ETYPES * NHEAD  * sizeof(float);
    unsigned* gmax = (unsigned*)ws;

    // zero accumulators (graph-capturable stream memsets)
    hipMemsetAsync(out,   0, (size_t)N_NODES * OUT_COLS * sizeof(float), stream);
    hipMemsetAsync(denom, 0, (size_t)N_NODES * NHEAD   * sizeof(float), stream);
    init_gmax_kernel<<<1, 1, 0, stream>>>(gmax);

    // 1) h = x @ W (WMMA f32)
    {
        const int strips = N_NODES / 16;                 // 3125
        const int wavesPerBlock = 8;                     // 256 threads
        const int blocks = (strips + wavesPerBlock - 1) / wavesPerBlock;
        gemm_h_kernel<<<blocks, 256, 0, stream>>>(x, W, h);
    }
    // 2) edge-type logits (tiny)
    etype_logits_kernel<<<1, 32, 0, stream>>>(edge_emb, W_e, a_e, he);
    // 3) per-node logits
    node_logits_kernel<<<(N_NODES * NHEAD + 255) / 256, 256, 0, stream>>>(h, a_l, a_r, hl, hr);
    // 4) edge scores + global max
    scores_kernel<<<(N_EDGES + 255) / 256, 256, 0, stream>>>(edge, hl, hr, he, scores, gmax);
    // 5) exp + denominator
    expsum_kernel<<<(N_EDGES + 255) / 256, 256, 0, stream>>>(edge, scores, gmax, denom);
    // 6) attention + aggregation (wave per edge)
    aggregate_kernel<<<((size_t)N_EDGES * 32 + 255) / 256, 256, 0, stream>>>(
        edge, h, scores, denom, out, attn_out);
}